// HashEmbedder_34557306864212
// MI455X (gfx1250) — compile-verified
//
#include <hip/hip_runtime.h>
#include <cstdint>
#include <cstddef>

// Flip to 0 to restore the plain (round-1) coordinate load path.
#define USE_ASYNC_X_STAGE 1

namespace {
constexpr int      kNLevels = 20;
constexpr int      kLog2T   = 19;
constexpr uint32_t kTMask   = (1u << kLog2T) - 1u;
constexpr int      kNPts    = 1000000;
constexpr int      kBlock   = 256;
constexpr int      kRowF2   = 23;             // float2 slots per point row (20 used + 3 pad)
constexpr uint32_t kPY      = 2654435761u;
constexpr uint32_t kPZ      = 805459861u;
}

// int(16 * 1.39**i) computed in double precision, truncated (matches Python)
__constant__ int kResTab[kNLevels] = {
    16, 22, 30, 42, 59, 83, 115, 160, 222, 309,
    430, 598, 832, 1157, 1608, 2235, 3107, 4319, 6003, 8345};

__global__ __launch_bounds__(kBlock)
void hashgrid_encode_kernel(const float* __restrict__ xin,
                            const float* __restrict__ tables,
                            float* __restrict__ out)
{
    __shared__ float2 stage[kBlock][kRowF2];   // 47,104 B
#if USE_ASYNC_X_STAGE
    __shared__ float  xbuf[kBlock * 3];        // 3,072 B coordinate tile
#endif

    const int t = threadIdx.x;
    const long long blockBase = (long long)blockIdx.x * kBlock;
    int p = (int)blockBase + t;
    if (p >= kNPts) p = kNPts - 1;             // tail lanes duplicate work; stores guarded

    float px, py, pz;

#if USE_ASYNC_X_STAGE
    {
        // Async DMA of this block's contiguous 768-float coordinate slice into LDS.
        // GVS form: 64-bit SGPR base + 32-bit per-lane byte offset (absolute, clamped
        // so the tail block never reads past the 12 MB allocation).
        const uint64_t sbase   = (uint64_t)(uintptr_t)xin;
        const uint32_t ldsBase = (uint32_t)(uintptr_t)&xbuf[0];   // flat->LDS: low 32 bits
        const uint32_t gfirst  = (uint32_t)(blockBase * 3);
        #pragma unroll
        for (int k = 0; k < 3; ++k) {
            const uint32_t elemLocal  = (uint32_t)t + 256u * (uint32_t)k;
            uint32_t       elemGlobal = gfirst + elemLocal;
            if (elemGlobal > 2999999u) elemGlobal = 2999999u;     // stay in bounds
            const uint32_t voff = elemGlobal * 4u;
            const uint32_t vlds = ldsBase + elemLocal * 4u;
            asm volatile("global_load_async_to_lds_b32 %0, %1, %2"
                         :: "v"(vlds), "v"(voff), "s"(sbase)
                         : "memory");
        }
#if __has_builtin(__builtin_amdgcn_s_wait_asynccnt)
        __builtin_amdgcn_s_wait_asynccnt(0);
#else
        asm volatile("s_wait_asynccnt 0x0" ::: "memory");
#endif
        __syncthreads();                        // make LDS tile visible to all waves
        const int pl = p - (int)blockBase;      // clamped local point index
        px = xbuf[pl * 3 + 0];
        py = xbuf[pl * 3 + 1];
        pz = xbuf[pl * 3 + 2];
    }
#else
    __builtin_prefetch(xin + (size_t)p * 3, 0, 3);   // near-cache prefetch
    px = xin[(size_t)p * 3 + 0];
    py = xin[(size_t)p * 3 + 1];
    pz = xin[(size_t)p * 3 + 2];
#endif

    for (int l = 0; l < kNLevels; ++l) {
        const float res = (float)kResTab[l];
        const float* __restrict__ tab = tables + ((size_t)l << (kLog2T + 1)); // l * T * F

        const float xs = px * res;
        const float ys = py * res;
        const float zs = pz * res;
        const float fx = floorf(xs), fy = floorf(ys), fz = floorf(zs);
        const float wx = xs - fx,    wy = ys - fy,    wz = zs - fz;

        const uint32_t hx0 = (uint32_t)(int)fx;
        const uint32_t hx1 = hx0 + 1u;
        const uint32_t hy0 = (uint32_t)(int)fy * kPY;
        const uint32_t hy1 = hy0 + kPY;
        const uint32_t hz0 = (uint32_t)(int)fz * kPZ;
        const uint32_t hz1 = hz0 + kPZ;

        // corner n = i*4 + j*2 + k  (i->x, j->y, k->z), matches reference OFFSETS order
        const uint32_t h0 = (hx0 ^ hy0 ^ hz0) & kTMask;
        const uint32_t h1 = (hx0 ^ hy0 ^ hz1) & kTMask;
        const uint32_t h2 = (hx0 ^ hy1 ^ hz0) & kTMask;
        const uint32_t h3 = (hx0 ^ hy1 ^ hz1) & kTMask;
        const uint32_t h4 = (hx1 ^ hy0 ^ hz0) & kTMask;
        const uint32_t h5 = (hx1 ^ hy0 ^ hz1) & kTMask;
        const uint32_t h6 = (hx1 ^ hy1 ^ hz0) & kTMask;
        const uint32_t h7 = (hx1 ^ hy1 ^ hz1) & kTMask;

        // 8 independent 8-byte gathers (global_load_b64), table row = exactly one float2
        const float2 e0 = *(const float2*)(tab + 2u * h0);
        const float2 e1 = *(const float2*)(tab + 2u * h1);
        const float2 e2 = *(const float2*)(tab + 2u * h2);
        const float2 e3 = *(const float2*)(tab + 2u * h3);
        const float2 e4 = *(const float2*)(tab + 2u * h4);
        const float2 e5 = *(const float2*)(tab + 2u * h5);
        const float2 e6 = *(const float2*)(tab + 2u * h6);
        const float2 e7 = *(const float2*)(tab + 2u * h7);

        // trilinear interpolation, fp32 (reference precision)
        const float c00x = fmaf(wx, e4.x - e0.x, e0.x);
        const float c00y = fmaf(wx, e4.y - e0.y, e0.y);
        const float c01x = fmaf(wx, e5.x - e1.x, e1.x);
        const float c01y = fmaf(wx, e5.y - e1.y, e1.y);
        const float c10x = fmaf(wx, e6.x - e2.x, e2.x);
        const float c10y = fmaf(wx, e6.y - e2.y, e2.y);
        const float c11x = fmaf(wx, e7.x - e3.x, e3.x);
        const float c11y = fmaf(wx, e7.y - e3.y, e3.y);
        const float c0x  = fmaf(wy, c10x - c00x, c00x);
        const float c0y  = fmaf(wy, c10y - c00y, c00y);
        const float c1x  = fmaf(wy, c11x - c01x, c01x);
        const float c1y  = fmaf(wy, c11y - c01y, c01y);

        stage[t][l] = make_float2(fmaf(wz, c1x - c0x, c0x),
                                  fmaf(wz, c1y - c0y, c0y));
    }

    __syncthreads();

    // Coalesced write-out: block's 256x40 floats leave as float4 stores,
    // consecutive lanes -> consecutive 16B -> 512B per wave per store.
    float4* __restrict__ out4 = (float4*)out;
    const long long out4Base = blockBase * 10;         // (blockBase*40)/4
    #pragma unroll
    for (int i = 0; i < 10; ++i) {
        const int m2  = i * 512 + t * 2;               // float2 index within block
        const int m2b = m2 + 1;
        const int p0 = m2  / 20, j0 = m2  - p0 * 20;
        const int p1 = m2b / 20, j1 = m2b - p1 * 20;
        const float2 a = stage[p0][j0];
        const float2 b = stage[p1][j1];
        const long long g4 = out4Base + i * 256 + t;
        if (g4 < (long long)kNPts * 10)
            out4[g4] = make_float4(a.x, a.y, b.x, b.y);
    }
}

extern "C" void kernel_launch(void* const* d_in, const int* in_sizes, int n_in,
                              void* d_out, int out_size, void* d_ws, size_t ws_size,
                              hipStream_t stream) {
    const float* x      = (const float*)d_in[0];   // (N_POINTS, 3) f32
    const float* tables = (const float*)d_in[1];   // (20, 2^19, 2) f32
    float* out          = (float*)d_out;           // (N_POINTS, 40) f32

    const int blocks = (kNPts + kBlock - 1) / kBlock;  // 3907
    hashgrid_encode_kernel<<<blocks, kBlock, 0, stream>>>(x, tables, out);
}